// ProxTV_23665269801384
// MI455X (gfx1250) — compile-verified
//
#include <hip/hip_runtime.h>

// ---------------------------------------------------------------------------
// Batched exact 1-D TV prox (Condat 2013) for MI455X / gfx1250.
//   B=4096 rows, N=2048 fp32, lambda scalar.
// One lane per row (wave32). Each block stages 32 rows into LDS (padded row
// stride to kill bank conflicts) via CDNA5 async global->LDS copies
// (ASYNCcnt path), runs the serial Condat state machine per lane with all
// per-iteration loads hitting LDS (software-pipelined one load ahead),
// overwrites the staged signal in place with the solution, and streams it
// back with async LDS->global stores.
// ---------------------------------------------------------------------------

namespace {
constexpr int kN       = 2048;          // signal length
constexpr int kL       = kN - 1;
constexpr int kRows    = 32;            // rows per block == wave32 lanes
constexpr int kStride  = kN + 1;        // LDS row stride (floats); +1 pad breaks
                                        // the 2048*4B = 64-bank alignment
constexpr int kMaxIter = 10 * kN;       // same safety cap as reference
constexpr int kChunks  = (kRows * kN) / (kRows * 4);  // 512 b128 copies / lane
}

// ---- CDNA5 async global<->LDS helpers (ASYNCcnt path), with safe fallback --

typedef int v4i __attribute__((__vector_size__(16)));
typedef __attribute__((address_space(1))) v4i* gbl_v4i_p;   // global (AS1)
typedef __attribute__((address_space(3))) v4i* lds_v4i_p;   // LDS    (AS3)

__device__ __forceinline__ void g2l_b128_async(const float* gsrc, float* ldst) {
#if __has_builtin(__builtin_amdgcn_global_load_async_to_lds_b128)
  __builtin_amdgcn_global_load_async_to_lds_b128(
      (gbl_v4i_p)gsrc, (lds_v4i_p)ldst, 0, 0);
#else
  *(float4*)ldst = *(const float4*)gsrc;   // synchronous fallback
#endif
}

__device__ __forceinline__ void l2g_b128_async(float* gdst, const float* lsrc) {
#if __has_builtin(__builtin_amdgcn_global_store_async_from_lds_b128)
  __builtin_amdgcn_global_store_async_from_lds_b128(
      (gbl_v4i_p)gdst, (lds_v4i_p)lsrc, 0, 0);
#else
  *(float4*)gdst = *(const float4*)lsrc;   // synchronous fallback
#endif
}

__device__ __forceinline__ void wait_async0() {
#if __has_builtin(__builtin_amdgcn_s_wait_asynccnt)
  __builtin_amdgcn_s_wait_asynccnt(0);
#else
  asm volatile("s_wait_asynccnt 0" ::: "memory");
#endif
}

__device__ __forceinline__ void wait_ds0() {
  // make this wave's ds_stores visible before the async engine reads LDS
  asm volatile("s_wait_dscnt 0" ::: "memory");
}

// ---------------------------------------------------------------------------

extern "C" __global__ __launch_bounds__(kRows)
void tv1d_condat_kernel(const float* __restrict__ x,
                        const float* __restrict__ lbda,
                        float* __restrict__ out) {
  extern __shared__ float lds[];        // kRows * kStride floats (~256.1 KB)

  const int   lane = threadIdx.x;       // 0..31, one row per lane
  const long long row0 = (long long)blockIdx.x * kRows;
  const float* gx   = x   + row0 * kN;
  float*       gout = out + row0 * kN;

  // ---- Stage 32 rows into LDS with async b128 copies (padded layout) ------
  for (int c = 0; c < kChunks; ++c) {
    const int f   = (c * kRows + lane) * 4;   // flat float idx in block tile
    const int r   = f >> 11;                  // f / kN
    const int col = f & (kN - 1);             // f % kN (4-float chunk, same row)
    g2l_b128_async(gx + f, &lds[r * kStride + col]);
  }
  wait_async0();                              // single wave: no barrier needed

  const float lam = *lbda;
  float* Y = &lds[lane * kStride];            // this lane's signal; solution is
                                              // written over it in place

  // ---- Condat direct algorithm (mirrors the reference state machine) ------
  // Invariant: whenever k < kL at the top of an iteration, ynext == Y[k+1].
  int   k = 0, k0 = 0, km = 0, kp = 0;
  const float y0 = Y[0];
  float ynext = Y[1];
  float vmin = y0 - lam, vmax = y0 + lam;
  float umin = lam, umax = -lam;
  bool  fresh = false, done = false;

  for (int it = 0; it < kMaxIter && !done; ++it) {
    if (k < kL) {                                   // main steps c/d/e
      const float yk1 = ynext;
      // Speculative load for the common no-jump path: issued before the
      // branch condition consumes yk1, so its LDS latency overlaps the VALU
      // work of this iteration. Clamped; only consumed when valid.
      const int   pf  = (k + 2 < kN) ? (k + 2) : kL;
      const float ypf = Y[pf];
      if (yk1 + umin < vmin - lam) {                // B: negative jump
        for (int i = k0; i <= km; ++i) Y[i] = vmin; // emit segment [k0..km]
        const int r = km + 1;                       // r <= kL
        const int rn = (r + 1 < kN) ? (r + 1) : kL;
        const float yB = Y[r];                      // two independent loads:
        ynext = Y[rn];                              // latencies overlap
        k = k0 = km = kp = r;
        vmin = yB;  vmax = yB + 2.0f * lam;
        umin = lam; umax = -lam;
      } else if (yk1 + umax > vmax + lam) {         // C: positive jump
        for (int i = k0; i <= kp; ++i) Y[i] = vmax; // emit segment [k0..kp]
        const int r = kp + 1;                       // r <= kL
        const int rn = (r + 1 < kN) ? (r + 1) : kL;
        const float yC = Y[r];
        ynext = Y[rn];
        k = k0 = km = kp = r;
        vmin = yC - 2.0f * lam; vmax = yC;
        umin = lam;             umax = -lam;
      } else {                                      // D: extend segment
        const int   kD = k + 1;
        const float u0 = umin + yk1 - vmin;
        const float w0 = umax + yk1 - vmax;
        const float seglen = (float)(kD - k0 + 1);
        if (u0 >= lam)  { vmin += (u0 - lam) / seglen; umin = lam;  km = kD; }
        else            { umin = u0; }
        if (w0 <= -lam) { vmax += (w0 + lam) / seglen; umax = -lam; kp = kD; }
        else            { umax = w0; }
        k = kD;
        ynext = ypf;                                // Y[k+1] for new k
      }
      fresh = false;
    } else if (fresh) {                             // A: lone trailing sample
      const float v = vmin + umin;
      for (int i = k0; i <= kL; ++i) Y[i] = v;
      done = true;
    } else if (umin < 0.0f) {                       // E: terminal neg jump
      for (int i = k0; i <= km; ++i) Y[i] = vmin;
      const int r = km + 1;                         // km < kL here
      const int rn = (r + 1 < kN) ? (r + 1) : kL;
      const float yB = Y[r];
      ynext = Y[rn];                                // Y[k+1] if next iter main
      k = k0 = km = r;                              // kp unchanged
      umax = yB + lam - vmax;                       // uses old vmax
      vmin = yB; umin = lam;
      fresh = true;
    } else if (umax > 0.0f) {                       // F: terminal pos jump
      for (int i = k0; i <= kp; ++i) Y[i] = vmax;
      const int r = kp + 1;                         // kp < kL here
      const int rn = (r + 1 < kN) ? (r + 1) : kL;
      const float yC = Y[r];
      ynext = Y[rn];
      k = k0 = kp = r;                              // km unchanged
      umin = yC - lam - vmin;                       // uses old vmin
      vmax = yC; umax = -lam;
      fresh = true;
    } else {                                        // G: global termination
      const float v = vmin + umin / (float)(k - k0 + 1);
      for (int i = k0; i <= kL; ++i) Y[i] = v;
      done = true;
    }
  }

  // ---- Stream the solution back, coalesced, via async LDS->global ---------
  wait_ds0();
  for (int c = 0; c < kChunks; ++c) {
    const int f   = (c * kRows + lane) * 4;
    const int r   = f >> 11;
    const int col = f & (kN - 1);
    l2g_b128_async(gout + f, &lds[r * kStride + col]);
  }
  wait_async0();                                    // drain before endpgm
}

// ---------------------------------------------------------------------------

extern "C" void kernel_launch(void* const* d_in, const int* in_sizes, int n_in,
                              void* d_out, int out_size, void* d_ws, size_t ws_size,
                              hipStream_t stream) {
  (void)n_in; (void)d_ws; (void)ws_size; (void)out_size;

  const float* x    = (const float*)d_in[0];   // (B, N) fp32
  const float* lbda = (const float*)d_in[1];   // scalar fp32
  float*       out  = (float*)d_out;           // (B, N) fp32

  const int total   = in_sizes[0];             // B * N
  const int nBlocks = total / (kRows * kN);    // 4096/32 = 128 workgroups

  const size_t ldsBytes = (size_t)kRows * kStride * sizeof(float); // 262,272 B
  // CDNA5 allows up to 320 KB LDS per workgroup; opt this kernel in.
  (void)hipFuncSetAttribute((const void*)tv1d_condat_kernel,
                            hipFuncAttributeMaxDynamicSharedMemorySize,
                            (int)ldsBytes);

  tv1d_condat_kernel<<<nBlocks, kRows, ldsBytes, stream>>>(x, lbda, out);
}